// VectorQuantizer_87282325389771
// MI455X (gfx1250) — compile-verified
//
#include <hip/hip_runtime.h>
#include <stdint.h>

// Problem constants (match reference)
constexpr int B_ = 8, T_ = 2048, D_ = 256, K_ = 8192;
constexpr int N_ = B_ * T_;                  // 16384 query vectors
constexpr int ROWS = 32;                     // query rows per block (2 WMMA M tiles)
constexpr int WAVES = 8;                     // wave32 x 8 = 256 threads
constexpr int THREADS = 256;
constexpr int CODES_PER_ITER = WAVES * 16;   // 128 codes scored per iteration
constexpr int ITERS = K_ / CODES_PER_ITER;   // 64
constexpr int DC = 16;                       // D floats staged per chunk
constexpr int DCHUNKS = D_ / DC;             // 16
constexpr int NCHUNK = ITERS * DCHUNKS;      // 1024 pipelined chunks
constexpr int XST = D_ + 4;                  // padded LDS stride (conflict-free)
constexpr int CST = DC + 4;                  // padded LDS stride for code chunk

typedef float v2f __attribute__((ext_vector_type(2)));
typedef float v8f __attribute__((ext_vector_type(8)));
typedef int vq_i4 __attribute__((__vector_size__(4 * sizeof(int))));

// CDNA5 async global->LDS staging (ASYNCcnt), with sync fallback
#if __has_builtin(__builtin_amdgcn_global_load_async_to_lds_b128) && \
    __has_builtin(__builtin_amdgcn_s_wait_asynccnt)
#define VQ_ASYNC 1
#else
#define VQ_ASYNC 0
#endif

#if VQ_ASYNC
__device__ __forceinline__ void async_b128(const float* g, float* l) {
  __builtin_amdgcn_global_load_async_to_lds_b128(
      (vq_i4*)g, (vq_i4*)l, /*offset=*/0, /*cpol=*/0);
}
#endif

// ---------------------------------------------------------------------------
// Kernel 1: codebook row norms ||e_k||^2  (one wave per row)
// ---------------------------------------------------------------------------
__global__ __launch_bounds__(256) void vq_norms(const float* __restrict__ cb,
                                                float* __restrict__ norms) {
  const int wave = threadIdx.x >> 5;
  const int lane = threadIdx.x & 31;
  const int row = blockIdx.x * 8 + wave;
  const float4* p = (const float4*)(cb + (size_t)row * D_);
  float s = 0.f;
#pragma unroll
  for (int c = 0; c < 2; ++c) {
    float4 v = p[lane * 2 + c];
    s += v.x * v.x + v.y * v.y + v.z * v.z + v.w * v.w;
  }
#pragma unroll
  for (int off = 16; off; off >>= 1) s += __shfl_xor(s, off, 32);
  if (lane == 0) norms[row] = s;
}

// ---------------------------------------------------------------------------
// Kernel 2: fp32 WMMA distances, argmin, gather, loss partial
// ---------------------------------------------------------------------------
__global__ __launch_bounds__(THREADS) void vq_main(const float* __restrict__ x,
                                                   const float* __restrict__ cb,
                                                   const float* __restrict__ norms,
                                                   float* __restrict__ outQ,
                                                   float* __restrict__ outIdx,
                                                   float* __restrict__ partial) {
  __shared__ float xs[ROWS * XST];                 // query tile, resident all kernel
  __shared__ float cs[2][CODES_PER_ITER * CST];    // double-buffered code chunks
  __shared__ float redV[WAVES][ROWS];
  __shared__ int   redI[WAVES][ROWS];
  __shared__ int   rowIdx[ROWS];
  __shared__ float redL[THREADS];

  const int tid = threadIdx.x;
  const int wave = tid >> 5;
  const int lane = tid & 31;
  const int rowBase = blockIdx.x * ROWS;

  // ---- stage X tile: 32 rows x 256 floats (async if available) ----
  {
    const int r = tid >> 3;              // 0..31
    const int c0 = (tid & 7) * 32;       // 0..224
    const float* src = x + (size_t)(rowBase + r) * D_ + c0;
    float* dst = &xs[r * XST + c0];
#if VQ_ASYNC
#pragma unroll
    for (int c = 0; c < 8; ++c) async_b128(src + c * 4, dst + c * 4);
#else
    float4 t[8];
#pragma unroll
    for (int c = 0; c < 8; ++c) t[c] = *(const float4*)(src + c * 4);
#pragma unroll
    for (int c = 0; c < 8; ++c) *(float4*)(dst + c * 4) = t[c];
#endif
  }

  // per-thread codebook staging offsets: 2 x b128 per chunk (128 rows x 16 floats)
  int roff[2], loff[2];
#pragma unroll
  for (int c = 0; c < 2; ++c) {
    const int f = c * THREADS + tid;     // 0..511 float4 slots
    const int r = f >> 2;                // 4 float4 per 16-float row -> row 0..127
    const int c4 = (f & 3) * 4;          // 0..12
    roff[c] = r * D_ + c4;               // global float offset within chunk slab
    loff[c] = r * CST + c4;              // LDS float offset within buffer
  }
#if !VQ_ASYNC
  float4 pend[2];
#endif

  // ---- prologue: issue chunk 0 into buffer 0 ----
#if VQ_ASYNC
#pragma unroll
  for (int c = 0; c < 2; ++c) async_b128(cb + roff[c], &cs[0][loff[c]]);
#else
#pragma unroll
  for (int c = 0; c < 2; ++c) pend[c] = *(const float4*)(cb + roff[c]);
#endif

  const int sub = lane & 15;             // WMMA m (A) / n (B)
  const int ksel = (lane >> 4) << 1;     // which k-pair this lane holds
  const float* xrow0 = &xs[sub * XST + ksel];          // A tile rows 0..15
  const float* xrow1 = &xs[(16 + sub) * XST + ksel];   // A tile rows 16..31
  const int crowOff = (wave * 16 + sub) * CST + ksel;  // B fragment base

  float bestV[16];
  int bestI[16];
#pragma unroll
  for (int j = 0; j < 16; ++j) { bestV[j] = 3.4e38f; bestI[j] = 0x7fffffff; }

  int buf = 0;
  for (int it = 0; it < ITERS; ++it) {
    const int kblock = it * CODES_PER_ITER;
    v8f acc0 = {0.f, 0.f, 0.f, 0.f, 0.f, 0.f, 0.f, 0.f};
    v8f acc1 = {0.f, 0.f, 0.f, 0.f, 0.f, 0.f, 0.f, 0.f};

    for (int dcIdx = 0; dcIdx < DCHUNKS; ++dcIdx) {
      // ---- commit pending chunk into cs[buf] ----
#if VQ_ASYNC
      __builtin_amdgcn_s_wait_asynccnt(0);   // this wave's async writes landed
#else
#pragma unroll
      for (int c = 0; c < 2; ++c) *(float4*)&cs[buf][loff[c]] = pend[c];
#endif
      __syncthreads();  // everyone's chunk data visible; old buffer fully consumed

      // ---- issue next chunk into the other buffer (overlaps compute) ----
      const int g = it * DCHUNKS + dcIdx;
      if (g + 1 < NCHUNK) {
        const int nit = (g + 1) >> 4;
        const int ndc = ((g + 1) & 15) * DC;
        const int cbase = nit * (CODES_PER_ITER * D_) + ndc;
#if VQ_ASYNC
#pragma unroll
        for (int c = 0; c < 2; ++c) async_b128(cb + cbase + roff[c], &cs[buf ^ 1][loff[c]]);
#else
#pragma unroll
        for (int c = 0; c < 2; ++c) pend[c] = *(const float4*)(cb + cbase + roff[c]);
#endif
      }

      // ---- fp32 WMMA: one B fragment feeds both M tiles ----
      const int dcBase = dcIdx * DC;
      const float* bp = &cs[buf][crowOff];
#pragma unroll
      for (int d = 0; d < DC; d += 4) {
        v2f a0 = *(const v2f*)(xrow0 + dcBase + d);
        v2f a1 = *(const v2f*)(xrow1 + dcBase + d);
        v2f b = *(const v2f*)(bp + d);
        acc0 = __builtin_amdgcn_wmma_f32_16x16x4_f32(false, a0, false, b,
                                                     (short)0, acc0, false, false);
        acc1 = __builtin_amdgcn_wmma_f32_16x16x4_f32(false, a1, false, b,
                                                     (short)0, acc1, false, false);
      }
      buf ^= 1;
    }

    // ---- fold scores into running per-row argmin ----
    const int cidx = kblock + wave * 16 + sub;
    const float cn = norms[cidx];
#pragma unroll
    for (int j = 0; j < 8; ++j) {
      const float s0 = cn - 2.0f * acc0[j];
      if (s0 < bestV[j] || (s0 == bestV[j] && cidx < bestI[j])) {
        bestV[j] = s0; bestI[j] = cidx;
      }
      const float s1 = cn - 2.0f * acc1[j];
      if (s1 < bestV[j + 8] || (s1 == bestV[j + 8] && cidx < bestI[j + 8])) {
        bestV[j + 8] = s1; bestI[j + 8] = cidx;
      }
    }
  }

  // ---- reduce across the 16 lanes of each half (columns of the tile) ----
#pragma unroll
  for (int j = 0; j < 16; ++j) {
    float v = bestV[j];
    int id = bestI[j];
#pragma unroll
    for (int off = 8; off; off >>= 1) {
      const float ov = __shfl_xor(v, off, 32);
      const int oi = __shfl_xor(id, off, 32);
      if (ov < v || (ov == v && oi < id)) { v = ov; id = oi; }
    }
    const int row = (j < 8) ? j : (j + 8);   // acc0 -> rows 0..15, acc1 -> 16..31
    if (lane == 0)  { redV[wave][row] = v;     redI[wave][row] = id; }
    if (lane == 16) { redV[wave][row + 8] = v; redI[wave][row + 8] = id; }
  }
  __syncthreads();

  // ---- combine across waves; write indices ----
  if (tid < ROWS) {
    float v = redV[0][tid];
    int id = redI[0][tid];
#pragma unroll
    for (int w = 1; w < WAVES; ++w) {
      const float ov = redV[w][tid];
      const int oi = redI[w][tid];
      if (ov < v || (ov == v && oi < id)) { v = ov; id = oi; }
    }
    rowIdx[tid] = id;
    outIdx[rowBase + tid] = (float)id;
  }
  __syncthreads();

  // ---- gather codebook rows, write quantized, accumulate loss partial ----
  float lsum = 0.f;
  {
    const int r = tid >> 3;
    const int c0 = (tid & 7) * 32;
    const int id = rowIdx[r];
    const float4* src = (const float4*)(cb + (size_t)id * D_ + c0);
    float4* dst = (float4*)(outQ + (size_t)(rowBase + r) * D_ + c0);
#pragma unroll
    for (int c = 0; c < 8; ++c) {
      float4 q = src[c];
      float4 xv = *(const float4*)&xs[r * XST + c0 + c * 4];
      dst[c] = q;
      float dx = q.x - xv.x, dy = q.y - xv.y, dz = q.z - xv.z, dw = q.w - xv.w;
      lsum += dx * dx + dy * dy + dz * dz + dw * dw;
    }
  }
  redL[tid] = lsum;
  __syncthreads();
  if (tid == 0) {
    float s = 0.f;
    for (int i = 0; i < THREADS; ++i) s += redL[i];
    partial[blockIdx.x] = s;  // deterministic per-block partial (no atomics)
  }
}

// ---------------------------------------------------------------------------
// Kernel 3: deterministic serial sum of loss partials
// ---------------------------------------------------------------------------
__global__ void vq_finalize(const float* __restrict__ partial,
                            float* __restrict__ outLoss, int nPartial) {
  if (threadIdx.x == 0 && blockIdx.x == 0) {
    float s = 0.f;
    for (int i = 0; i < nPartial; ++i) s += partial[i];
    outLoss[0] = 0.25f * s / (float)((size_t)N_ * D_);
  }
}

// ---------------------------------------------------------------------------
extern "C" void kernel_launch(void* const* d_in, const int* in_sizes, int n_in,
                              void* d_out, int out_size, void* d_ws, size_t ws_size,
                              hipStream_t stream) {
  const float* x  = (const float*)d_in[0];   // (B,T,D) fp32
  const float* cb = (const float*)d_in[1];   // (K,D) fp32

  float* outQ    = (float*)d_out;            // N*D quantized
  float* outIdx  = outQ + (size_t)N_ * D_;   // N indices (as float)
  float* outLoss = outIdx + N_;              // 1 scalar

  float* norms   = (float*)d_ws;             // K floats
  float* partial = norms + K_;               // N/ROWS floats

  vq_norms<<<K_ / 8, 256, 0, stream>>>(cb, norms);
  vq_main<<<N_ / ROWS, THREADS, 0, stream>>>(x, cb, norms, outQ, outIdx, partial);
  vq_finalize<<<1, 32, 0, stream>>>(partial, outLoss, N_ / ROWS);
}